// PartitionPadding_9105330668092
// MI455X (gfx1250) — compile-verified
//
#include <hip/hip_runtime.h>
#include <stdint.h>

// PartitionPadding on MI455X (gfx1250):
//   ragged [N=65536, D=512] f32 rows -> dense [B=512, max_atoms, D] zero-padded,
//   implemented with the CDNA5 Tensor Data Mover: OOB reads return zero, which
//   performs the padding in hardware. TENSORcnt tracks completion.

typedef unsigned int v4u __attribute__((ext_vector_type(4)));
typedef int          v8i __attribute__((ext_vector_type(8)));
typedef int          v4i __attribute__((ext_vector_type(4)));

#define PP_BATCH 512
#define PP_DFEAT 512
#define PP_CHROWS 32   // rows per TDM tile: 32 * 512 * 4B = 64 KB LDS

// ---------------------------------------------------------------------------
// Kernel 1: starts[m] = lower_bound(molecule_indicator, m), starts[B] = N.
// molecule_indicator is sorted, values in [0, B-1].
// ---------------------------------------------------------------------------
__global__ void pp_starts_kernel(const int* __restrict__ ind, int n,
                                 int* __restrict__ starts) {
    int m = blockIdx.x * blockDim.x + threadIdx.x;
    if (m > PP_BATCH) return;
    int lo = 0, hi = n;
    while (lo < hi) {
        int mid = (lo + hi) >> 1;
        if (ind[mid] < m) lo = mid + 1; else hi = mid;
    }
    starts[m] = lo;
}

// ---------------------------------------------------------------------------
// TDM descriptor builder (D# groups 0 and 1), per CDNA5 ISA:
//   group0: [1:0]=count=1, [63:32]=lds_addr, [120:64]=global_addr, [127:126]=type=2
//   group1: [17:16]=data_size(2 -> 4B), [79:48]=tensor_dim0, [111:80]=tensor_dim1,
//           [127:112]=tile_dim0, [143:128]=tile_dim1, [207:160]=tensor_dim0_stride
// 2-D tensor: groups 2/3 are zero, tile_dim2 = 0, dim1_stride unused.
// ---------------------------------------------------------------------------
static __device__ inline void pp_tdm_desc(v4u& g0, v8i& g1,
                                          unsigned lds_addr,
                                          unsigned long long gaddr,
                                          int tensor_d1, int tile_d1) {
    g0[0] = 1u;                                                   // count=1, user D#
    g0[1] = lds_addr;                                             // LDS byte address
    g0[2] = (unsigned)(gaddr & 0xFFFFFFFFull);                    // global_addr[31:0]
    g0[3] = (unsigned)((gaddr >> 32) & 0x01FFFFFFull) | (2u << 30); // [56:32] | type=2

    g1[0] = (2 << 16);                                            // data_size = 4 bytes
    g1[1] = (int)((PP_DFEAT & 0xFFFF) << 16);                     // tensor_dim0[15:0]
    g1[2] = (int)((PP_DFEAT >> 16) & 0xFFFF) |
            (int)((tensor_d1 & 0xFFFF) << 16);                    // dim0 hi | dim1 lo
    g1[3] = (int)((tensor_d1 >> 16) & 0xFFFF) |
            (int)(PP_DFEAT << 16);                                // dim1 hi | tile_dim0
    g1[4] = (int)(tile_d1 & 0xFFFF);                              // tile_dim1, tile_dim2=0
    g1[5] = PP_DFEAT;                                             // tensor_dim0_stride lo
    g1[6] = 0;                                                    // stride0 hi | dim1_stride lo
    g1[7] = 0;                                                    // dim1_stride hi
}

// ---------------------------------------------------------------------------
// Kernel 2: one wave per (molecule b, 32-row chunk c).
//   TDM load : src rows [s0+p0, s0+cnt) with tensor_dim1 = cnt-p0 -> rows past
//              the molecule read as ZERO (hardware padding) into LDS.
//   TDM store: dense tile -> out[b, p0:p0+tile_rows, :].
// ---------------------------------------------------------------------------
__launch_bounds__(32)
__global__ void pp_pad_tdm_kernel(const float* __restrict__ atom,
                                  float* __restrict__ out,
                                  const int* __restrict__ starts,
                                  int max_atoms, int nchunks) {
    __shared__ float lds_tile[PP_CHROWS * PP_DFEAT];   // 64 KB

    const int wg = blockIdx.x;
    const int b  = wg / nchunks;
    const int c  = wg - b * nchunks;
    const int p0 = c * PP_CHROWS;

    int s0  = starts[b];
    int cnt = starts[b + 1] - s0;

    int tile_rows = max_atoms - p0; if (tile_rows > PP_CHROWS) tile_rows = PP_CHROWS;
    int src_rows  = cnt - p0;       if (src_rows < 0) src_rows = 0;

    // Descriptor words must be wave-uniform (SGPRs).
    s0        = __builtin_amdgcn_readfirstlane(s0);
    tile_rows = __builtin_amdgcn_readfirstlane(tile_rows);
    src_rows  = __builtin_amdgcn_readfirstlane(src_rows);

    // Flat shared pointer: low 32 bits are the LDS byte offset (ISA 10.2 aperture map).
    const unsigned lds_base = (unsigned)(uintptr_t)(void*)lds_tile;

    const unsigned long long src_addr =
        (unsigned long long)(uintptr_t)(atom + (size_t)(s0 + p0) * PP_DFEAT);
    const unsigned long long dst_addr =
        (unsigned long long)(uintptr_t)(out + ((size_t)b * (size_t)max_atoms + (size_t)p0) * PP_DFEAT);

    v4u g0; v8i g1;
    v4i gz4 = {0, 0, 0, 0};
    v8i gz8 = {0, 0, 0, 0, 0, 0, 0, 0};

    // Global -> LDS: tensor_dim1 = remaining valid rows; OOB rows load as zero.
    pp_tdm_desc(g0, g1, lds_base, src_addr, src_rows, tile_rows);
    __builtin_amdgcn_tensor_load_to_lds(g0, g1, gz4, gz4, gz8, 0);
    __builtin_amdgcn_s_wait_tensorcnt(0);

    // LDS -> Global: full tile in range, nothing dropped.
    pp_tdm_desc(g0, g1, lds_base, dst_addr, tile_rows, tile_rows);
    __builtin_amdgcn_tensor_store_from_lds(g0, g1, gz4, gz4, gz8, 0);
    __builtin_amdgcn_s_wait_tensorcnt(0);
}

// ---------------------------------------------------------------------------
extern "C" void kernel_launch(void* const* d_in, const int* in_sizes, int n_in,
                              void* d_out, int out_size, void* d_ws, size_t ws_size,
                              hipStream_t stream) {
    const float* atom = (const float*)d_in[0];   // [N, 512] f32
    const int*   ind  = (const int*)d_in[1];     // [N] i32, sorted segment ids
    float*       out  = (float*)d_out;           // [B, max_atoms, 512] f32

    const int n         = in_sizes[1];                       // N = 65536
    int*      starts    = (int*)d_ws;                        // B+1 ints of scratch
    const int max_atoms = out_size / (PP_BATCH * PP_DFEAT);  // all molecules kept
    const int nchunks   = (max_atoms + PP_CHROWS - 1) / PP_CHROWS;

    pp_starts_kernel<<<(PP_BATCH + 1 + 255) / 256, 256, 0, stream>>>(ind, n, starts);
    pp_pad_tdm_kernel<<<PP_BATCH * nchunks, 32, 0, stream>>>(atom, out, starts,
                                                             max_atoms, nchunks);
}